// StructAttention_52613349376835
// MI455X (gfx1250) — compile-verified
//
#include <hip/hip_runtime.h>

typedef __attribute__((ext_vector_type(2))) float v2f;
typedef __attribute__((ext_vector_type(8))) float v8f;

#define BB 32
#define NN_TOK 512
#define DD 768
#define HH 1024

#define WMMA_F32(acc, a, b) \
  __builtin_amdgcn_wmma_f32_16x16x4_f32(false, (a), false, (b), (short)0, (acc), false, false)

// ---------------------------------------------------------------------------
// Tiled fp32 WMMA GEMM.
//   C[m,n] = epi( sum_k A[m,k] * B'[k,n] )
//   BT_FROM_NT=true : B given as (N,K) row-major (we compute A @ B^T, straight copy to LDS)
//   BT_FROM_NT=false: B given as (K,N) row-major (A @ B, transpose-store into LDS)
//   EPI: 0 = tanh(c + bias[n]);  1 = identity;  2 = (m==n) ? 0 : exp(c)
// Tile: BM=128, BN=64, BK=32; 256 threads = 8 waves (4x2), wave -> 32x32 (2x2 of 16x16).
// ---------------------------------------------------------------------------
template<int EPI, bool BT_FROM_NT>
__global__ __launch_bounds__(256) void gemm_wmma_f32(
    const float* __restrict__ Ag, const float* __restrict__ Bg,
    const float* __restrict__ bias, float* __restrict__ Cg,
    int K, int lda, int ldb, int ldc,
    size_t aStride, size_t bStride, size_t cStride)
{
  constexpr int BM = 128, BN = 64, BK = 32, LK = BK + 4;   // stride 36: 16B-aligned rows, conflict-free frags
  __shared__ float As[BM * LK];
  __shared__ float Bs[BN * LK];

  const int z = blockIdx.z;
  const float* A = Ag + (size_t)z * aStride;
  const float* Bsrc = Bg + (size_t)z * bStride;
  float* C = Cg + (size_t)z * cStride;

  const int bm = blockIdx.y * BM, bn = blockIdx.x * BN;
  const int tid = threadIdx.x, lane = tid & 31, wave = tid >> 5;
  const int wm = (wave & 3) * 32, wn = (wave >> 2) * 32;

  v8f acc[2][2] = {};

  for (int k0 = 0; k0 < K; k0 += BK) {
    // --- A tile: 128x32 floats, 4 float4 per thread
#pragma unroll
    for (int i = 0; i < 4; ++i) {
      int slot = tid + i * 256;
      int r = slot >> 3, ks = (slot & 7) << 2;
      float4 g = *(const float4*)(A + (size_t)(bm + r) * lda + k0 + ks);
      *(float4*)(&As[r * LK + ks]) = g;
    }
    // --- B tile -> Bt[n][k]
    if (BT_FROM_NT) {
#pragma unroll
      for (int i = 0; i < 2; ++i) {
        int slot = tid + i * 256;
        int r = slot >> 3, ks = (slot & 7) << 2;
        float4 g = *(const float4*)(Bsrc + (size_t)(bn + r) * ldb + k0 + ks);
        *(float4*)(&Bs[r * LK + ks]) = g;
      }
    } else {
#pragma unroll
      for (int i = 0; i < 2; ++i) {
        int slot = tid + i * 256;                 // 32 k-rows x 16 n4-groups
        int kk = slot >> 4, n4 = (slot & 15) << 2;
        float4 g = *(const float4*)(Bsrc + (size_t)(k0 + kk) * ldb + bn + n4);
        Bs[(n4 + 0) * LK + kk] = g.x;
        Bs[(n4 + 1) * LK + kk] = g.y;
        Bs[(n4 + 2) * LK + kk] = g.z;
        Bs[(n4 + 3) * LK + kk] = g.w;
      }
    }
    __syncthreads();

    const int fr = lane & 15, fk2 = (lane >> 4) << 1;
#pragma unroll
    for (int ks = 0; ks < BK; ks += 4) {
      v2f a[2], b[2];
#pragma unroll
      for (int mi = 0; mi < 2; ++mi) {
        const float* p = &As[(wm + mi * 16 + fr) * LK + ks + fk2];
        a[mi].x = p[0]; a[mi].y = p[1];
      }
#pragma unroll
      for (int ni = 0; ni < 2; ++ni) {
        const float* p = &Bs[(wn + ni * 16 + fr) * LK + ks + fk2];
        b[ni].x = p[0]; b[ni].y = p[1];
      }
#pragma unroll
      for (int mi = 0; mi < 2; ++mi)
#pragma unroll
        for (int ni = 0; ni < 2; ++ni)
          acc[mi][ni] = WMMA_F32(acc[mi][ni], a[mi], b[ni]);
    }
    __syncthreads();
  }

  // --- epilogue + store (C/D layout: M = r + 8*(lane/16), N = lane%16)
  const int cn = lane & 15, rhalf = (lane >> 4) << 3;
#pragma unroll
  for (int mi = 0; mi < 2; ++mi)
#pragma unroll
    for (int ni = 0; ni < 2; ++ni) {
      int gm0 = bm + wm + mi * 16 + rhalf;
      int gn = bn + wn + ni * 16 + cn;
#pragma unroll
      for (int r = 0; r < 8; ++r) {
        int gm = gm0 + r;
        float v = acc[mi][ni][r];
        if (EPI == 0)      v = tanhf(v + bias[gn]);
        else if (EPI == 2) v = (gm == gn) ? 0.0f : expf(v);
        C[(size_t)gm * ldc + gn] = v;
      }
    }
}

// ---------------------------------------------------------------------------
// Dual-GEMM output kernel (per batch z):
//   U = P @ Linv        (NN: transpose-store tiles of Linv)
//   V = P @ Linv^T      (NT: straight-copy tiles of Linv)
//   out[i,j] = (j>0 ? U[i,j] : 0) - (i>0 ? V[i,j] : 0)
// Shares the P tile between both accumulator sets.
// ---------------------------------------------------------------------------
__global__ __launch_bounds__(256) void gemm_dual_out(
    const float* __restrict__ Pg, const float* __restrict__ Lg, float* __restrict__ Og)
{
  constexpr int BM = 128, BN = 64, BK = 32, LK = BK + 4;
  __shared__ float As[BM * LK];
  __shared__ float Bn[BN * LK];   // Bn[n][k] = Linv[k][n]
  __shared__ float Bt[BN * LK];   // Bt[n][k] = Linv[n][k]

  const int z = blockIdx.z;
  const float* A = Pg + (size_t)z * NN_TOK * NN_TOK;
  const float* L = Lg + (size_t)z * NN_TOK * NN_TOK;
  float* O = Og + (size_t)z * NN_TOK * NN_TOK;
  const int ld = NN_TOK, K = NN_TOK;

  const int bm = blockIdx.y * BM, bn = blockIdx.x * BN;
  const int tid = threadIdx.x, lane = tid & 31, wave = tid >> 5;
  const int wm = (wave & 3) * 32, wn = (wave >> 2) * 32;

  v8f accU[2][2] = {}, accV[2][2] = {};

  for (int k0 = 0; k0 < K; k0 += BK) {
#pragma unroll
    for (int i = 0; i < 4; ++i) {
      int slot = tid + i * 256;
      int r = slot >> 3, ks = (slot & 7) << 2;
      float4 g = *(const float4*)(A + (size_t)(bm + r) * ld + k0 + ks);
      *(float4*)(&As[r * LK + ks]) = g;
    }
#pragma unroll
    for (int i = 0; i < 2; ++i) {             // NT (straight): Bt[n][k] = Linv[bn+n][k0+k]
      int slot = tid + i * 256;
      int r = slot >> 3, ks = (slot & 7) << 2;
      float4 g = *(const float4*)(L + (size_t)(bn + r) * ld + k0 + ks);
      *(float4*)(&Bt[r * LK + ks]) = g;
    }
#pragma unroll
    for (int i = 0; i < 2; ++i) {             // NN (transpose-store): Bn[n][k] = Linv[k0+k][bn+n]
      int slot = tid + i * 256;
      int kk = slot >> 4, n4 = (slot & 15) << 2;
      float4 g = *(const float4*)(L + (size_t)(k0 + kk) * ld + bn + n4);
      Bn[(n4 + 0) * LK + kk] = g.x;
      Bn[(n4 + 1) * LK + kk] = g.y;
      Bn[(n4 + 2) * LK + kk] = g.z;
      Bn[(n4 + 3) * LK + kk] = g.w;
    }
    __syncthreads();

    const int fr = lane & 15, fk2 = (lane >> 4) << 1;
#pragma unroll
    for (int ks = 0; ks < BK; ks += 4) {
      v2f a[2], bu[2], bv[2];
#pragma unroll
      for (int mi = 0; mi < 2; ++mi) {
        const float* p = &As[(wm + mi * 16 + fr) * LK + ks + fk2];
        a[mi].x = p[0]; a[mi].y = p[1];
      }
#pragma unroll
      for (int ni = 0; ni < 2; ++ni) {
        const float* pu = &Bn[(wn + ni * 16 + fr) * LK + ks + fk2];
        bu[ni].x = pu[0]; bu[ni].y = pu[1];
        const float* pv = &Bt[(wn + ni * 16 + fr) * LK + ks + fk2];
        bv[ni].x = pv[0]; bv[ni].y = pv[1];
      }
#pragma unroll
      for (int mi = 0; mi < 2; ++mi)
#pragma unroll
        for (int ni = 0; ni < 2; ++ni) {
          accU[mi][ni] = WMMA_F32(accU[mi][ni], a[mi], bu[ni]);
          accV[mi][ni] = WMMA_F32(accV[mi][ni], a[mi], bv[ni]);
        }
    }
    __syncthreads();
  }

  const int cn = lane & 15, rhalf = (lane >> 4) << 3;
#pragma unroll
  for (int mi = 0; mi < 2; ++mi)
#pragma unroll
    for (int ni = 0; ni < 2; ++ni) {
      int gm0 = bm + wm + mi * 16 + rhalf;
      int gn = bn + wn + ni * 16 + cn;
#pragma unroll
      for (int r = 0; r < 8; ++r) {
        int gm = gm0 + r;
        float u = accU[mi][ni][r];
        float v = accV[mi][ni][r];
        float o = (gn > 0 ? u : 0.0f) - (gm > 0 ? v : 0.0f);
        O[(size_t)gm * ld + gn] = o;
      }
    }
}

// ---------------------------------------------------------------------------
// rexp[b] = exp( x[b,0,:] . W_R + b_R )
// ---------------------------------------------------------------------------
__global__ void rexp_kernel(const float* __restrict__ x, const float* __restrict__ W_R,
                            const float* __restrict__ b_R, float* __restrict__ rexp)
{
  const int b = blockIdx.x;
  const float* xb = x + (size_t)b * NN_TOK * DD;   // token 0 row
  float s = 0.0f;
  for (int d = threadIdx.x; d < DD; d += 256) s += xb[d] * W_R[d];
  __shared__ float red[256];
  red[threadIdx.x] = s;
  __syncthreads();
  for (int off = 128; off > 0; off >>= 1) {
    if (threadIdx.x < off) red[threadIdx.x] += red[threadIdx.x + off];
    __syncthreads();
  }
  if (threadIdx.x == 0) rexp[b] = expf(red[0] + b_R[0]);
}

// ---------------------------------------------------------------------------
// Build L_var from P (column-per-thread; P has zero diagonal already):
//   L[i][j] = -P[i][j];  L[j][j] = colsum_j;  L[0][j] = rexp[b]
// ---------------------------------------------------------------------------
__global__ void build_L_kernel(const float* __restrict__ P, const float* __restrict__ rexp,
                               float* __restrict__ Lout)
{
  const int b = blockIdx.x;
  const float* Pb = P + (size_t)b * NN_TOK * NN_TOK;
  float* Lb = Lout + (size_t)b * NN_TOK * NN_TOK;
  const int j = threadIdx.x;
  float s = 0.0f;
  for (int i = 0; i < NN_TOK; ++i) {
    float p = Pb[(size_t)i * NN_TOK + j];
    s += p;
    Lb[(size_t)i * NN_TOK + j] = -p;
  }
  Lb[(size_t)j * NN_TOK + j] = s;       // diagonal = column sum
  Lb[j] = rexp[b];                      // row 0 override (incl. j==0)
}

// ---------------------------------------------------------------------------
// In-place Gauss-Jordan inversion (no pivoting), one 512-thread WG per matrix.
// Each thread owns one column; pivot row & pivot column staged in LDS.
// ---------------------------------------------------------------------------
__global__ __launch_bounds__(512) void gj_inverse_kernel(float* __restrict__ Lbase)
{
  float* M = Lbase + (size_t)blockIdx.x * NN_TOK * NN_TOK;
  const int c = threadIdx.x;
  __shared__ float prow[NN_TOK];
  __shared__ float fcol[NN_TOK];

  for (int k = 0; k < NN_TOK; ++k) {
    float mkc = M[(size_t)k * NN_TOK + c];
    prow[c] = mkc;
    fcol[c] = M[(size_t)c * NN_TOK + k];
    __syncthreads();

    float d = 1.0f / prow[k];
    float pk = (c == k) ? d : mkc * d;     // scaled pivot row (implicit identity trick)
    M[(size_t)k * NN_TOK + c] = pk;

    for (int i = 0; i < NN_TOK; ++i) {
      if (i == k) continue;
      float f = fcol[i];
      float v = M[(size_t)i * NN_TOK + c];
      v = (c == k) ? (-f * d) : fmaf(-f, pk, v);
      M[(size_t)i * NN_TOK + c] = v;
    }
    __syncthreads();
  }
}

// ---------------------------------------------------------------------------
extern "C" void kernel_launch(void* const* d_in, const int* in_sizes, int n_in,
                              void* d_out, int out_size, void* d_ws, size_t ws_size,
                              hipStream_t stream)
{
  (void)in_sizes; (void)n_in; (void)out_size; (void)ws_size;

  const float* x    = (const float*)d_in[0];   // (B,N,D)
  const float* W_P  = (const float*)d_in[1];   // (H,D)
  const float* b_P  = (const float*)d_in[2];   // (H)
  const float* W_C  = (const float*)d_in[3];   // (H,D)
  const float* b_C  = (const float*)d_in[4];   // (H)
  const float* W_bl = (const float*)d_in[5];   // (H,H)
  const float* W_R  = (const float*)d_in[6];   // (1,D)
  const float* b_R  = (const float*)d_in[7];   // (1)
  float* out = (float*)d_out;                  // (B,N,N)

  float* ws = (float*)d_ws;
  const size_t szEnc = (size_t)BB * NN_TOK * HH;   // 16.78M floats
  float* encP = ws;                 // [0, szEnc)
  float* encC = ws + szEnc;         // [szEnc, 2szEnc)
  float* T    = ws + 2 * szEnc;     // [2szEnc, 3szEnc)
  float* P    = encP;               // reuse: encP dead after T
  float* L    = encC;               // reuse: encC dead after pw
  float* rexp = ws + 3 * szEnc;     // 32 floats

  const int MTOK = BB * NN_TOK;     // 16384
  dim3 blk(256);

  // enc_P = tanh(x @ W_P^T + b_P); enc_C likewise
  gemm_wmma_f32<0, true><<<dim3(HH / 64, MTOK / 128, 1), blk, 0, stream>>>(
      x, W_P, b_P, encP, DD, DD, DD, HH, 0, 0, 0);
  gemm_wmma_f32<0, true><<<dim3(HH / 64, MTOK / 128, 1), blk, 0, stream>>>(
      x, W_C, b_C, encC, DD, DD, DD, HH, 0, 0, 0);

  // T = enc_P @ W_bl
  gemm_wmma_f32<1, false><<<dim3(HH / 64, MTOK / 128, 1), blk, 0, stream>>>(
      encP, W_bl, nullptr, T, HH, HH, HH, HH, 0, 0, 0);

  // P[b] = offdiag * exp( T[b] @ enc_C[b]^T )
  gemm_wmma_f32<2, true><<<dim3(NN_TOK / 64, NN_TOK / 128, BB), blk, 0, stream>>>(
      T, encC, nullptr, P, HH, HH, HH, NN_TOK,
      (size_t)NN_TOK * HH, (size_t)NN_TOK * HH, (size_t)NN_TOK * NN_TOK);

  // root scores
  rexp_kernel<<<BB, 256, 0, stream>>>(x, W_R, b_R, rexp);

  // Laplacian with row-0 override, then in-place inverse
  build_L_kernel<<<BB, NN_TOK, 0, stream>>>(P, rexp, L);
  gj_inverse_kernel<<<BB, NN_TOK, 0, stream>>>(L);

  // A = mask_j * (P @ Linv) - mask_i * (P @ Linv^T)
  gemm_dual_out<<<dim3(NN_TOK / 64, NN_TOK / 128, BB), blk, 0, stream>>>(P, L, out);
}